// GraphIsomorphismLayer_22643067584730
// MI455X (gfx1250) — compile-verified
//
#include <hip/hip_runtime.h>

typedef __attribute__((ext_vector_type(2))) float v2f;
typedef __attribute__((ext_vector_type(8))) float v8f;

#define TS 130  // padded LDS row stride in dwords (kills 16-way bank conflicts on stride-128)

// ---------------- init: pooled = (1+eps)*h ; zero counters & stats ----------------
__global__ __launch_bounds__(256) void k_init(const float* __restrict__ h,
                                              const float* __restrict__ epsp,
                                              float* __restrict__ pooled,
                                              int* __restrict__ cnt,
                                              float* __restrict__ stats,
                                              int N) {
  int i = blockIdx.x * blockDim.x + threadIdx.x;
  float c = 1.0f + *epsp;
  int n4 = N * 32;  // N*128/4 float4 elements
  if (i < n4) {
    float4 v = ((const float4*)h)[i];
    v.x *= c; v.y *= c; v.z *= c; v.w *= c;
    ((float4*)pooled)[i] = v;
  }
  if (i < N) cnt[i] = 0;
  if (i < 512) stats[i] = 0.0f;
}

// ---------------- degree histogram ----------------
__global__ __launch_bounds__(256) void k_hist(const int* __restrict__ dst,
                                              int* __restrict__ cnt, int E) {
  int e = blockIdx.x * blockDim.x + threadIdx.x;
  if (e < E) atomicAdd(&cnt[dst[e]], 1);
}

// ---------------- two-level exclusive scan (chunk = 1024, supports N <= 262144) ----------------
__global__ __launch_bounds__(256) void k_scan_blocks(const int* __restrict__ cnt,
                                                     int* __restrict__ offs,
                                                     int* __restrict__ bsums, int N) {
  __shared__ int sh[256];
  int t = threadIdx.x;
  int base = blockIdx.x * 1024 + t * 4;
  int v0 = (base + 0 < N) ? cnt[base + 0] : 0;
  int v1 = (base + 1 < N) ? cnt[base + 1] : 0;
  int v2 = (base + 2 < N) ? cnt[base + 2] : 0;
  int v3 = (base + 3 < N) ? cnt[base + 3] : 0;
  int tsum = v0 + v1 + v2 + v3;
  sh[t] = tsum;
  __syncthreads();
  for (int d = 1; d < 256; d <<= 1) {
    int x = (t >= d) ? sh[t - d] : 0;
    __syncthreads();
    sh[t] += x;
    __syncthreads();
  }
  if (t == 255) bsums[blockIdx.x] = sh[255];
  int run = sh[t] - tsum;  // exclusive prefix of this thread within block
  if (base + 0 < N) offs[base + 0] = run; run += v0;
  if (base + 1 < N) offs[base + 1] = run; run += v1;
  if (base + 2 < N) offs[base + 2] = run; run += v2;
  if (base + 3 < N) offs[base + 3] = run;
}

__global__ __launch_bounds__(256) void k_scan_sums(int* __restrict__ bsums, int nb) {
  __shared__ int sh[256];
  int t = threadIdx.x;
  int v = (t < nb) ? bsums[t] : 0;
  sh[t] = v;
  __syncthreads();
  for (int d = 1; d < 256; d <<= 1) {
    int x = (t >= d) ? sh[t - d] : 0;
    __syncthreads();
    sh[t] += x;
    __syncthreads();
  }
  if (t < nb) bsums[t] = sh[t] - v;  // exclusive
}

__global__ __launch_bounds__(256) void k_scan_add(int* __restrict__ offs,
                                                  const int* __restrict__ bsums,
                                                  int* __restrict__ cursor, int N) {
  int i = blockIdx.x * blockDim.x + threadIdx.x;
  if (i < N) {
    int o = offs[i] + bsums[i >> 10];
    offs[i] = o;
    cursor[i] = o;
  }
}

// ---------------- counting-sort scatter of src by dst ----------------
__global__ __launch_bounds__(256) void k_scatter(const int* __restrict__ src,
                                                 const int* __restrict__ dst,
                                                 int* __restrict__ cursor,
                                                 int* __restrict__ sorted, int E) {
  int e = blockIdx.x * blockDim.x + threadIdx.x;
  if (e < E) {
    int p = atomicAdd(&cursor[dst[e]], 1);
    sorted[p] = src[e];
  }
}

// ---------------- segmented neighbor sum: one wave per node, no float atomics ----------------
__global__ __launch_bounds__(256) void k_aggregate(const float* __restrict__ h,
                                                   const int* __restrict__ srt,
                                                   const int* __restrict__ offs,
                                                   float* __restrict__ pooled,
                                                   int N, int E) {
  int w = (blockIdx.x * blockDim.x + threadIdx.x) >> 5;  // wave32: one wave per node
  int lane = threadIdx.x & 31;
  if (w >= N) return;
  int beg = offs[w];
  int end = (w + 1 < N) ? offs[w + 1] : E;
  size_t lo = (size_t)lane * 4;  // lane owns dwords [4l..4l+3] -> 512B coalesced per row
  float4 acc = *(const float4*)(pooled + (size_t)w * 128 + lo);
  int j = beg;
  for (; j + 1 < end; j += 2) {  // 2x unroll to overlap the two row fetches
    int s0 = srt[j], s1 = srt[j + 1];
    float4 a = *(const float4*)(h + (size_t)s0 * 128 + lo);
    float4 b = *(const float4*)(h + (size_t)s1 * 128 + lo);
    acc.x += a.x + b.x; acc.y += a.y + b.y; acc.z += a.z + b.z; acc.w += a.w + b.w;
  }
  if (j < end) {
    int s0 = srt[j];
    float4 a = *(const float4*)(h + (size_t)s0 * 128 + lo);
    acc.x += a.x; acc.y += a.y; acc.z += a.z; acc.w += a.w;
  }
  *(float4*)(pooled + (size_t)w * 128 + lo) = acc;
}

// ---------------- fused [optional BN+ReLU on input] GEMM (Kx128 @ 128x128) + bias + column stats ----------------
// Block: 256 threads = 8 waves; block tile 64 rows x 128 cols; wave w owns 16-col stripe, 4 row subtiles.
// Uses V_WMMA_F32_16X16X4_F32 (fp32 WMMA, 32 K-steps).
__global__ __launch_bounds__(256) void k_gemm_bn(const float* __restrict__ A,
                                                 const float* __restrict__ W,
                                                 const float* __restrict__ bias,
                                                 const float* __restrict__ in_scale,
                                                 const float* __restrict__ in_shift,
                                                 float* __restrict__ out,
                                                 float* __restrict__ colsum,
                                                 float* __restrict__ colsumsq,
                                                 int Nrows) {
  __shared__ float Wt[128 * TS];   // W transposed: Wt[n][k] = W[k][n]
  __shared__ float Al[64 * TS];    // A tile
  const int t = threadIdx.x;
  const int blockRow = blockIdx.x * 64;

  for (int i = t; i < 128 * 128; i += 256) {
    int k = i >> 7, n = i & 127;
    Wt[n * TS + k] = W[i];                 // coalesced global read, scattered LDS write (once per block)
  }
  for (int i = t; i < 64 * 128; i += 256) {
    int r = i >> 7, c = i & 127;
    int gr = blockRow + r;
    float v = 0.0f;
    if (gr < Nrows) {
      v = A[(size_t)gr * 128 + c];
      if (in_scale) v = fmaxf(fmaf(v, in_scale[c], in_shift[c]), 0.0f);  // fused BN+ReLU
    }
    Al[r * TS + c] = v;
  }
  __syncthreads();

  const int lane = t & 31;
  const int wv = t >> 5;        // 0..7 -> 16-col stripe
  const int lm = lane & 15;
  const int lh = lane >> 4;

  v8f acc0 = {}, acc1 = {}, acc2 = {}, acc3 = {};
  const float* bp = Wt + (16 * wv + lm) * TS + 2 * lh;   // B frag: {W[k][n], W[k+1][n]}
  const float* ap = Al + lm * TS + 2 * lh;               // A frag: {A[m][k], A[m][k+1]}

#pragma unroll 4
  for (int k0 = 0; k0 < 128; k0 += 4) {
    v2f b_ = *(const v2f*)(bp + k0);
    v2f a_;
    a_ = *(const v2f*)(ap + k0);
    acc0 = __builtin_amdgcn_wmma_f32_16x16x4_f32(false, a_, false, b_, (short)0, acc0, false, false);
    a_ = *(const v2f*)(ap + 16 * TS + k0);
    acc1 = __builtin_amdgcn_wmma_f32_16x16x4_f32(false, a_, false, b_, (short)0, acc1, false, false);
    a_ = *(const v2f*)(ap + 32 * TS + k0);
    acc2 = __builtin_amdgcn_wmma_f32_16x16x4_f32(false, a_, false, b_, (short)0, acc2, false, false);
    a_ = *(const v2f*)(ap + 48 * TS + k0);
    acc3 = __builtin_amdgcn_wmma_f32_16x16x4_f32(false, a_, false, b_, (short)0, acc3, false, false);
  }

  // C/D layout: lane holds column (16*wv + lm) for all 8 VGPRs; row = 16*mt + 8*lh + v
  const int col = 16 * wv + lm;
  const float bcol = bias[col];
  float s1 = 0.0f, s2 = 0.0f;

#define EMIT_FULL(ACC, MT)                                                   \
  {                                                                          \
    _Pragma("unroll") for (int v = 0; v < 8; ++v) {                          \
      int row = blockRow + 16 * (MT) + 8 * lh + v;                           \
      float val = ACC[v] + bcol;                                             \
      out[(size_t)row * 128 + col] = val;                                    \
      s1 += val;                                                             \
      s2 += val * val;                                                       \
    }                                                                        \
  }
#define EMIT_GUARD(ACC, MT)                                                  \
  {                                                                          \
    _Pragma("unroll") for (int v = 0; v < 8; ++v) {                          \
      int row = blockRow + 16 * (MT) + 8 * lh + v;                           \
      float val = ACC[v] + bcol;                                             \
      if (row < Nrows) {                                                     \
        out[(size_t)row * 128 + col] = val;                                  \
        s1 += val;                                                           \
        s2 += val * val;                                                     \
      }                                                                      \
    }                                                                        \
  }
  if (blockRow + 64 <= Nrows) {  // uniform: all but the last ragged block take the branch-free path
    EMIT_FULL(acc0, 0) EMIT_FULL(acc1, 1) EMIT_FULL(acc2, 2) EMIT_FULL(acc3, 3)
  } else {
    EMIT_GUARD(acc0, 0) EMIT_GUARD(acc1, 1) EMIT_GUARD(acc2, 2) EMIT_GUARD(acc3, 3)
  }
#undef EMIT_FULL
#undef EMIT_GUARD

  // lanes l and l+16 hold the same column -> pair-reduce, then 1 atomic/column/wave
  s1 += __shfl_xor(s1, 16);
  s2 += __shfl_xor(s2, 16);
  if (lh == 0) {
    atomicAdd(&colsum[col], s1);
    atomicAdd(&colsumsq[col], s2);
  }
}

// ---------------- BN stats -> per-column scale/shift ----------------
__global__ void k_bn_fin(const float* __restrict__ csum, const float* __restrict__ csq,
                         const float* __restrict__ g, const float* __restrict__ be,
                         float* __restrict__ scale, float* __restrict__ shift, int N) {
  int c = threadIdx.x;
  if (c < 128) {
    float inv = 1.0f / (float)N;
    float mu = csum[c] * inv;
    float var = csq[c] * inv - mu * mu;  // biased variance, matches reference
    float r = rsqrtf(var + 1e-5f);
    float sc = g[c] * r;
    scale[c] = sc;
    shift[c] = be[c] - mu * sc;
  }
}

// ---------------- final BN + ReLU -> d_out ----------------
__global__ __launch_bounds__(256) void k_final(const float* __restrict__ P,
                                               const float* __restrict__ scale,
                                               const float* __restrict__ shift,
                                               float* __restrict__ out, int N) {
  int i = blockIdx.x * blockDim.x + threadIdx.x;
  if (i < N * 32) {
    float4 x = ((const float4*)P)[i];
    int c4 = i & 31;
    float4 sc = ((const float4*)scale)[c4];
    float4 sf = ((const float4*)shift)[c4];
    float4 r;
    r.x = fmaxf(fmaf(x.x, sc.x, sf.x), 0.0f);
    r.y = fmaxf(fmaf(x.y, sc.y, sf.y), 0.0f);
    r.z = fmaxf(fmaf(x.z, sc.z, sf.z), 0.0f);
    r.w = fmaxf(fmaf(x.w, sc.w, sf.w), 0.0f);
    ((float4*)out)[i] = r;
  }
}

extern "C" void kernel_launch(void* const* d_in, const int* in_sizes, int n_in,
                              void* d_out, int out_size, void* d_ws, size_t ws_size,
                              hipStream_t stream) {
  const float* h   = (const float*)d_in[0];
  const int*   esrc = (const int*)d_in[1];
  const int*   edst = (const int*)d_in[2];
  const float* W1  = (const float*)d_in[3];
  const float* b1  = (const float*)d_in[4];
  const float* g1  = (const float*)d_in[5];
  const float* be1 = (const float*)d_in[6];
  const float* W2  = (const float*)d_in[7];
  const float* b2  = (const float*)d_in[8];
  const float* g2  = (const float*)d_in[9];
  const float* be2 = (const float*)d_in[10];
  const float* eps = (const float*)d_in[11];

  const int N = in_sizes[0] / 128;
  const int E = in_sizes[1];

  auto align16 = [](size_t x) { return (x + 15) & ~(size_t)15; };
  char* ws = (char*)d_ws;
  size_t off = 0;
  float* pooled = (float*)(ws + off); off = align16(off + (size_t)N * 128 * sizeof(float));
  int*   sorted = (int*)(ws + off);   off = align16(off + (size_t)E * sizeof(int));
  int*   offs   = (int*)(ws + off);   off = align16(off + (size_t)N * sizeof(int));
  int*   cursor = (int*)(ws + off);   off = align16(off + (size_t)N * sizeof(int));
  int*   cnt    = (int*)(ws + off);   off = align16(off + (size_t)N * sizeof(int));
  int*   bsums  = (int*)(ws + off);   off = align16(off + 256 * sizeof(int));
  float* stats  = (float*)(ws + off); off = align16(off + 1024 * sizeof(float));
  float* colsum1 = stats;       float* colsq1 = stats + 128;
  float* colsum2 = stats + 256; float* colsq2 = stats + 384;
  float* scale1  = stats + 512; float* shift1 = stats + 640;
  float* scale2  = stats + 768; float* shift2 = stats + 896;
  float* X = (float*)d_out;  // x1 lives in d_out temporarily (consumed before final write)

  const int tInit = N * 32;
  k_init<<<(tInit + 255) / 256, 256, 0, stream>>>(h, eps, pooled, cnt, stats, N);
  k_hist<<<(E + 255) / 256, 256, 0, stream>>>(edst, cnt, E);
  const int nb = (N + 1023) / 1024;
  k_scan_blocks<<<nb, 256, 0, stream>>>(cnt, offs, bsums, N);
  k_scan_sums<<<1, 256, 0, stream>>>(bsums, nb);
  k_scan_add<<<(N + 255) / 256, 256, 0, stream>>>(offs, bsums, cursor, N);
  k_scatter<<<(E + 255) / 256, 256, 0, stream>>>(esrc, edst, cursor, sorted, E);
  k_aggregate<<<(N * 32 + 255) / 256, 256, 0, stream>>>(h, sorted, offs, pooled, N, E);
  const int gb = (N + 63) / 64;
  k_gemm_bn<<<gb, 256, 0, stream>>>(pooled, W1, b1, nullptr, nullptr, X, colsum1, colsq1, N);
  k_bn_fin<<<1, 128, 0, stream>>>(colsum1, colsq1, g1, be1, scale1, shift1, N);
  k_gemm_bn<<<gb, 256, 0, stream>>>(X, W2, b2, scale1, shift1, pooled, colsum2, colsq2, N);
  k_bn_fin<<<1, 128, 0, stream>>>(colsum2, colsq2, g2, be2, scale2, shift2, N);
  k_final<<<(tInit + 255) / 256, 256, 0, stream>>>(pooled, scale2, shift2, (float*)d_out, N);
}